// DistributionQuantizedCDFTable_46196668236283
// MI455X (gfx1250) — compile-verified
//
#include <hip/hip_runtime.h>
#include <stdint.h>

#define WAVES_PER_BLOCK 8
#define ML 64      // max_length (fixed by problem shape)
#define NCDF 66    // max_length + 2

// Truncate generic flat pointer to 32 bits: per CDNA5 ISA, flat LDS addresses
// are {SHARED_BASE, lds_offset[31:0]}, so the low 32 bits are the wave-relative
// LDS byte offset expected by global_load_async_to_lds_* VDST.
__device__ __forceinline__ uint32_t lds_off32(const void* p) {
    return (uint32_t)(uintptr_t)p;
}

__device__ __forceinline__ float wred_addf(float v) {
#pragma unroll
    for (int m = 16; m > 0; m >>= 1) v += __shfl_xor(v, m, 32);
    return v;
}
__device__ __forceinline__ int wred_addi(int v) {
#pragma unroll
    for (int m = 16; m > 0; m >>= 1) v += __shfl_xor(v, m, 32);
    return v;
}
__device__ __forceinline__ int wred_mini(int v) {
#pragma unroll
    for (int m = 16; m > 0; m >>= 1) { int t = __shfl_xor(v, m, 32); v = t < v ? t : v; }
    return v;
}
// Exclusive prefix sum across the wave; also returns the wave total.
__device__ __forceinline__ int wscan_excl(int v, int lane, int* tot) {
    int s = v;
#pragma unroll
    for (int d = 1; d < 32; d <<= 1) {
        int t = __shfl_up(s, d, 32);
        if (lane >= d) s += t;
    }
    *tot = __shfl(s, 31, 32);
    return s - v;
}

__global__ __launch_bounds__(WAVES_PER_BLOCK * 32)
void quantized_cdf_kernel(const float* __restrict__ pmf,
                          const int* __restrict__ pmf_length,
                          const int* __restrict__ precision_p,
                          int32_t* __restrict__ out, int C) {
    __shared__ float smem[WAVES_PER_BLOCK][ML];
    const int lane = threadIdx.x & 31;
    const int wave = threadIdx.x >> 5;
    const int c = blockIdx.x * WAVES_PER_BLOCK + wave;
    if (c >= C) return;

    const int L = pmf_length[c];          // wave-uniform, 8..64
    const int P = precision_p[0];         // 16
    const int scale = 1 << P;

    // ---- Stage this channel's pmf row into LDS via the CDNA5 async path ----
    {
        const float* g = pmf + (size_t)c * ML + lane * 2;       // 8B per lane, 256B per wave
        uint32_t lo = lds_off32(&smem[wave][lane * 2]);
        asm volatile("global_load_async_to_lds_b64 %0, %1, off"
                     :: "v"(lo), "v"(g) : "memory");
        asm volatile("s_wait_asynccnt 0" ::: "memory");
    }
    float p0 = smem[wave][lane];          // slot j = lane
    float p1 = smem[wave][lane + 32];     // slot j = lane+32

    // overflow = clip(1 - sum(p[j<L]), 0)
    float m0 = (lane < L) ? p0 : 0.0f;
    float m1 = (lane + 32 < L) ? p1 : 0.0f;
    float s = wred_addf(m0 + m1);
    float ovf = 1.0f - s;
    if (ovf < 0.0f) ovf = 0.0f;

    // pvec: pmf for j<L, overflow at j==L, else 0 (slot j=64 handled as uniform q2)
    float pv0 = (lane < L) ? p0 : ((lane == L) ? ovf : 0.0f);
    float pv1 = (lane + 32 < L) ? p1 : ((lane + 32 == L) ? ovf : 0.0f);

    // freq = floor(pvec_f64 * 2^P + 0.5), masked to j<=L  (matches ref f64 math)
    const double dscale = (double)scale;
    int q0 = (lane <= L)      ? (int)floor((double)pv0 * dscale + 0.5) : 0;
    int q1 = (lane + 32 <= L) ? (int)floor((double)pv1 * dscale + 0.5) : 0;
    int q2 = (L == ML)        ? (int)floor((double)ovf * dscale + 0.5) : 0;  // j = 64

    // renormalize: freq = (freq << P) / max(total,1)   (exact integer division)
    int total = wred_addi(q0 + q1) + q2;
    if (total < 1) total = 1;
    const uint64_t ut = (uint64_t)(uint32_t)total;
    q0 = (int)(((uint64_t)(uint32_t)q0 << P) / ut);
    q1 = (int)(((uint64_t)(uint32_t)q1 << P) / ut);
    q2 = (int)(((uint64_t)(uint32_t)q2 << P) / ut);

    // cdf = [0, cumsum(freq)];  c0=cdf[lane], c1=cdf[lane+32], c2=cdf[64+lane] (lanes 0,1)
    int T0, T1;
    int e0 = wscan_excl(q0, lane, &T0);
    int e1 = wscan_excl(q1, lane, &T1);
    int c0 = e0;
    int c1 = T0 + e1;
    int c2 = T0 + T1 + ((lane >= 1) ? q2 : 0);

    // cdf[L+1] = 2^P
    const int kp = L + 1;
    if (kp < 32)      { if (lane == kp)      c0 = scale; }
    else if (kp < 64) { if (lane == kp - 32) c1 = scale; }
    else              { if (lane == kp - 64) c2 = scale; }

    // ---- Sequential fixup loop (wave-uniform trip count; body wave-uniform) ----
    const int BIG = 0x7fffffff;
    for (int i = 0; i <= L; ++i) {
        int k = i;
        int cdfi  = (k < 32) ? __shfl(c0, k, 32)
                  : (k < 64) ? __shfl(c1, k - 32, 32)
                             : __shfl(c2, k - 64, 32);
        k = i + 1;
        int cdfi1 = (k < 32) ? __shfl(c0, k, 32)
                  : (k < 64) ? __shfl(c1, k - 32, 32)
                             : __shfl(c2, k - 64, 32);
        if (cdfi == cdfi1) {   // zero-width interval at i: steal from smallest f>1
            // interval widths f[j] = cdf[j+1] - cdf[j]
            int n0  = __shfl(c0, (lane + 1) & 31, 32);
            int b32 = __shfl(c1, 0, 32);   // cdf[32]
            int n1  = __shfl(c1, (lane + 1) & 31, 32);
            int b64 = __shfl(c2, 0, 32);   // cdf[64]
            int b65 = __shfl(c2, 1, 32);   // cdf[65]
            int f0   = ((lane == 31) ? b32 : n0) - c0;   // j = lane
            int f1   = ((lane == 31) ? b64 : n1) - c1;   // j = lane+32
            int f64v = b65 - b64;                        // j = 64 (uniform)

            // packed key keeps "first smallest" tie-break of the C++ scan
            int key0 = (lane <= L && f0 > 1)        ? ((f0 << 7) | lane)        : BIG;
            int key1 = ((lane + 32) <= L && f1 > 1) ? ((f1 << 7) | (lane + 32)) : BIG;
            int key  = key0 < key1 ? key0 : key1;
            key = wred_mini(key);
            int key2 = (L == ML && f64v > 1) ? ((f64v << 7) | 64) : BIG;
            key = key < key2 ? key : key2;
            int best = key & 127;

            if (best < i) {   // cdf[best+1 .. i] -= 1
                c0 -= (int)((lane      > best) & (lane      <= i));
                c1 -= (int)((lane + 32 > best) & (lane + 32 <= i));
                c2 -= (int)((lane + 64 > best) & (lane + 64 <= i));
            } else {          // cdf[i+1 .. best] += 1
                c0 += (int)((lane      > i) & (lane      <= best));
                c1 += (int)((lane + 32 > i) & (lane + 32 <= best));
                c2 += (int)((lane + 64 > i) & (lane + 64 <= best));
            }
        }
    }

    // zero-pad past L+1 and store
    int32_t* o = out + (size_t)c * NCDF;
    o[lane]      = (lane      <= kp) ? c0 : 0;
    o[lane + 32] = (lane + 32 <= kp) ? c1 : 0;
    if (lane < 2) o[64 + lane] = ((64 + lane) <= kp) ? c2 : 0;
}

extern "C" void kernel_launch(void* const* d_in, const int* in_sizes, int n_in,
                              void* d_out, int out_size, void* d_ws, size_t ws_size,
                              hipStream_t stream) {
    const float* pmf        = (const float*)d_in[0];
    const int*   pmf_length = (const int*)d_in[1];
    // d_in[2] = max_length (fixed 64 by layout), d_in[3] = precision
    const int*   precision  = (const int*)d_in[3];
    const int C = in_sizes[1];
    const int blocks = (C + WAVES_PER_BLOCK - 1) / WAVES_PER_BLOCK;
    hipLaunchKernelGGL(quantized_cdf_kernel, dim3(blocks), dim3(WAVES_PER_BLOCK * 32),
                       0, stream, pmf, pmf_length, precision, (int32_t*)d_out, C);
}